// WholeMask_63264868270544
// MI455X (gfx1250) — compile-verified
//
#include <hip/hip_runtime.h>
#include <cstdint>
#include <cstddef>

#define IMG      512
#define NB       8
#define NK       16
#define MH       28
#define MW       28
#define BAND     32
#define NTHREADS 256

typedef float v4f __attribute__((ext_vector_type(4)));

__global__ __launch_bounds__(NTHREADS) void paste_masks_kernel(
    const float* __restrict__ bboxess,   // (B,K,4)
    const int*   __restrict__ counts,    // (B,1)
    const float* __restrict__ maskss,    // (B,K,1,28,28)
    float*       __restrict__ out)       // (B,K,1,512,512)
{
    __shared__ float m[MH * MW];   // 3136 B mask tile
    __shared__ int   sxl[IMG];     // per-x source column (or -1 outside box)
    __shared__ int   syl[BAND];    // per-row LDS row base = sy*MW (or -1)

    const int t    = threadIdx.x;
    const int band = blockIdx.x & 15;   // 16 bands of 32 rows
    const int bk   = blockIdx.x >> 4;   // 0..127
    const int b    = bk >> 4;           // K = 16
    const int k    = bk & 15;

    // ---- bbox: round-half-even then clip, exactly like the reference ----
    const float* bb = bboxess + bk * 4;
    int bi0 = (int)__builtin_rintf(bb[0]);
    int bi1 = (int)__builtin_rintf(bb[1]);
    int bi2 = (int)__builtin_rintf(bb[2]);
    int bi3 = (int)__builtin_rintf(bb[3]);
    int y1 = min(max(bi0, 0), IMG - 1);
    int x1 = min(max(bi1, 0), IMG - 1);
    int y2 = min(max(bi2, y1 + 1), IMG);
    int x2 = min(max(bi3, x1 + 1), IMG);
    int h  = y2 - y1;
    int w  = x2 - x1;
    const bool valid = (k < counts[b]);

    const int  r0      = band * BAND;
    const bool bandHit = valid && (r0 < y2) && (r0 + BAND > y1);

    v4f* outv = (v4f*)(out + (size_t)bk * ((size_t)IMG * IMG));
    const int c4   = t & 127;   // which float4 in the row (128 per row)
    const int rsub = t >> 7;    // 2 rows per pass

    // ======== fast path: band entirely zero (uniform branch) ========
    if (!bandHit) {
        const v4f z = (v4f)0.0f;
#pragma unroll
        for (int p = 0; p < BAND / 2; ++p) {
            int r = p * 2 + rsub;
            __builtin_nontemporal_store(z, &outv[(size_t)(r0 + r) * (IMG / 4) + c4]);
        }
        return;
    }

    // ======== box path ========
    // CDNA5 async global->LDS staging of the 28x28 mask (196 lanes x b128)
    if (t < (MH * MW / 4)) {
        uint32_t lds_addr = (uint32_t)(uintptr_t)(&m[0]) + (uint32_t)(t * 16);
        const float* gsrc = maskss + (size_t)bk * (MH * MW) + t * 4;
        asm volatile("global_load_async_to_lds_b128 %0, %1, off"
                     :: "v"(lds_addr), "v"(gsrc)
                     : "memory");
    }

    // overlap LUT construction with the async copy
    for (int x = t; x < IMG; x += NTHREADS) {
        int s = -1;
        if (x >= x1 && x < x2) {
            int v = ((x - x1) * MW) / w;      // exact floor division, rel >= 0
            s = v > (MW - 1) ? (MW - 1) : v;
        }
        sxl[x] = s;
    }
    if (t < BAND) {
        int y = r0 + t;
        int s = -1;
        if (y >= y1 && y < y2) {
            int v = ((y - y1) * MH) / h;
            if (v > MH - 1) v = MH - 1;
            s = v * MW;                        // precomputed LDS row base
        }
        syl[t] = s;
    }

    // drain this wave's async copies, then make LDS visible workgroup-wide
    asm volatile("s_wait_asynccnt 0x0" ::: "memory");
    __syncthreads();

    // per-thread source columns are loop-invariant: load once
    const int xx = c4 * 4;
    const int s0 = sxl[xx + 0];
    const int s1 = sxl[xx + 1];
    const int s2 = sxl[xx + 2];
    const int s3 = sxl[xx + 3];

#pragma unroll
    for (int p = 0; p < BAND / 2; ++p) {
        int r       = p * 2 + rsub;
        int rowbase = syl[r];
        v4f v = (v4f)0.0f;
        if (rowbase >= 0) {
            v.x = (s0 >= 0) ? m[rowbase + s0] : 0.0f;
            v.y = (s1 >= 0) ? m[rowbase + s1] : 0.0f;
            v.z = (s2 >= 0) ? m[rowbase + s2] : 0.0f;
            v.w = (s3 >= 0) ? m[rowbase + s3] : 0.0f;
        }
        __builtin_nontemporal_store(v, &outv[(size_t)(r0 + r) * (IMG / 4) + c4]);
    }
}

extern "C" void kernel_launch(void* const* d_in, const int* in_sizes, int n_in,
                              void* d_out, int out_size, void* d_ws, size_t ws_size,
                              hipStream_t stream) {
    const float* bboxess = (const float*)d_in[0];
    const int*   counts  = (const int*)d_in[1];
    const float* maskss  = (const float*)d_in[2];
    float*       out     = (float*)d_out;

    dim3 grid(NB * NK * (IMG / BAND));   // 2048 blocks
    paste_masks_kernel<<<grid, NTHREADS, 0, stream>>>(bboxess, counts, maskss, out);
}